// GraphSAGEPlusPlusDAMC_12481174962731
// MI455X (gfx1250) — compile-verified
//
#include <hip/hip_runtime.h>
#include <math.h>

// ---------------------------------------------------------------------------
// GraphSAGE++ (mean+max SAGE convs, 2 edge sets) for MI455X / gfx1250.
// Dense GEMMs use V_WMMA_F32_16X16X4_F32 (fp32-exact, wave32).
// Weights are repacked as K-pairs so each WMMA B-fragment is one b64 load.
// Aggregations: chained wave-per-edge scatters with global_prefetch of the
// next gather row; feature table (20.5MB) is L2-resident (192MB L2).
// ---------------------------------------------------------------------------

typedef float v2f __attribute__((ext_vector_type(2)));
typedef float v8f __attribute__((ext_vector_type(8)));

#define DFEAT 128
#define HDIM  128
#define NCLS  64

// ---------------- elementwise helpers ----------------

__global__ void relu_kernel(const float* __restrict__ x, float* __restrict__ y, int n) {
    int i = blockIdx.x * 256 + threadIdx.x;
    if (i < n) y[i] = fmaxf(x[i], 0.0f);
}

// W: [128 n][128 k] row-major -> packed B: pair p=k>>1, float idx p*256 + n*2 + (k&1)
// so fragment (k,k+1)@n is a contiguous, 8B-aligned float2.
__global__ void pack_w128_kernel(const float* __restrict__ W, float* __restrict__ P) {
    int i = blockIdx.x * 256 + threadIdx.x;   // 0..16383
    int n = i >> 7, k = i & 127;
    P[(k >> 1) * 256 + n * 2 + (k & 1)] = W[i];
}

// post_W: [64 n][512 k] -> packed: float idx (k>>1)*128 + n*2 + (k&1)
__global__ void pack_post_kernel(const float* __restrict__ W, float* __restrict__ P) {
    int i = blockIdx.x * 256 + threadIdx.x;   // 0..32767
    int n = i >> 9, k = i & 511;
    P[(k >> 1) * 128 + n * 2 + (k & 1)] = W[i];
}

__global__ void init_logits_kernel(float* __restrict__ logits, const float* __restrict__ b, int n) {
    int i = blockIdx.x * 256 + threadIdx.x;
    if (i < n) logits[i] = b[i & (NCLS - 1)];
}

__global__ void fill_u32_kernel(unsigned* __restrict__ p, unsigned v, int n) {
    int i = blockIdx.x * 256 + threadIdx.x;
    if (i < n) p[i] = v;
}

// ---------------- order-preserving float <-> uint map for atomicMax ----------------

__device__ __forceinline__ unsigned f32_ord(float f) {
    unsigned u = __float_as_uint(f);
    return (u & 0x80000000u) ? ~u : (u | 0x80000000u);
}
__device__ __forceinline__ float ord_f32(unsigned u) {
    return (u & 0x80000000u) ? __uint_as_float(u & 0x7FFFFFFFu) : __uint_as_float(~u);
}
#define ORD_NEG_INF 0x007FFFFFu   // f32_ord(-inf)

// ---------------- scatter aggregations ----------------
// Chained wave-per-edge: uniform edge index per wave (src[e]/dst[e] scalarize
// to SMEM), lanes move 4 floats each, next row prefetched to hide L2 latency.

#define SCATTER_BLOCKS 4096

__global__ void __launch_bounds__(256)
scatter_mean_kernel(const int* __restrict__ src, const int* __restrict__ dst,
                    const float* __restrict__ feat, float* __restrict__ agg,
                    float* __restrict__ cnt, int E) {
    int lane = threadIdx.x & 31;
    int gw = blockIdx.x * 8 + (threadIdx.x >> 5);
    int stride = SCATTER_BLOCKS * 8;
    for (int e = gw; e < E; e += stride) {
        int s = src[e], d = dst[e];
        int en = e + stride;
        if (en < E) {   // prefetch next gather row (global_prefetch_b8)
            int sn = src[en];
            __builtin_prefetch(feat + (long)sn * DFEAT + lane * 4, 0, 0);
        }
        const float4* f4 = (const float4*)(feat + (long)s * DFEAT);
        float4 v = f4[lane];                           // 4 floats/lane = 128/wave
        float* ap = agg + (long)d * DFEAT + lane * 4;
        atomicAdd(ap + 0, v.x); atomicAdd(ap + 1, v.y);
        atomicAdd(ap + 2, v.z); atomicAdd(ap + 3, v.w);
        if (lane == 0) atomicAdd(cnt + d, 1.0f);
    }
}

__global__ void __launch_bounds__(256)
scatter_max_kernel(const int* __restrict__ src, const int* __restrict__ dst,
                   const float* __restrict__ feat, unsigned* __restrict__ agg, int E) {
    int lane = threadIdx.x & 31;
    int gw = blockIdx.x * 8 + (threadIdx.x >> 5);
    int stride = SCATTER_BLOCKS * 8;
    for (int e = gw; e < E; e += stride) {
        int s = src[e], d = dst[e];
        int en = e + stride;
        if (en < E) {
            int sn = src[en];
            __builtin_prefetch(feat + (long)sn * DFEAT + lane * 4, 0, 0);
        }
        const float4* f4 = (const float4*)(feat + (long)s * DFEAT);
        float4 v = f4[lane];
        unsigned* ap = agg + (long)d * DFEAT + lane * 4;
        atomicMax(ap + 0, f32_ord(v.x)); atomicMax(ap + 1, f32_ord(v.y));
        atomicMax(ap + 2, f32_ord(v.z)); atomicMax(ap + 3, f32_ord(v.w));
    }
}

__global__ void mean_div_kernel(float* __restrict__ agg, const float* __restrict__ cnt, int n) {
    int i = blockIdx.x * 256 + threadIdx.x;
    if (i < n) agg[i] = agg[i] / fmaxf(cnt[i >> 7], 1.0f);
}

// decode monotone-uint back to float; isolated targets (-inf) -> 0 (PyG fill)
__global__ void max_decode_kernel(const unsigned* in, float* outp, int n) {
    int i = blockIdx.x * 256 + threadIdx.x;
    if (i < n) {
        unsigned u = in[i];
        outp[i] = (u == ORD_NEG_INF) ? 0.0f : ord_f32(u);
    }
}

// ---------------- WMMA GEMMs ----------------
// Fragment conventions (ISA 7.12.2, wave32):
//   A 16x4 f32 : lane L -> M = L&15 ; a.x = K=(L<16?0:2), a.y = K+1
//   B 4x16 f32 : lane L -> N = L&15 ; b.x = K=(L<16?0:2), b.y = K+1
//   C/D 16x16  : vgpr r, lane L -> M = r + (L<16?0:8), N = L&15

// tmp[16-row strip][128] = agg @ Wl^T + h @ Wr^T + bias.  8 waves x 16-col tiles.
// Bl/Br are the packed weight layouts from pack_w128_kernel.
__global__ void __launch_bounds__(256)
gemm_conv_kernel(const float* __restrict__ Aagg, const float* __restrict__ Ah,
                 const float* __restrict__ Blp,  const float* __restrict__ Brp,
                 const float* __restrict__ bias, float* __restrict__ out) {
    __shared__ float lds[2 * 16 * 132];          // +4 pad: stride 132 -> conflict-free
    float* Sa = lds;
    float* Sh = lds + 16 * 132;
    int tid = threadIdx.x;
    long row0 = (long)blockIdx.x * 16;
    #pragma unroll
    for (int j = 0; j < 8; ++j) {                // 256 thr load 2x 16x128 coalesced
        int i = tid + j * 256;
        int r = i >> 7, c = i & 127;
        Sa[r * 132 + c] = Aagg[(row0 + r) * DFEAT + c];
        Sh[r * 132 + c] = Ah  [(row0 + r) * DFEAT + c];
    }
    __syncthreads();

    int lane = tid & 31, wave = tid >> 5;
    int m    = lane & 15;
    int koff = (lane < 16) ? 0 : 2;
    int kh   = (lane < 16) ? 0 : 1;              // K-pair half
    int n    = wave * 16 + m;
    const v2f* Bl2 = (const v2f*)Blp;            // float2 index: p*128 + n
    const v2f* Br2 = (const v2f*)Brp;

    v8f acc = {};
    #pragma unroll 4
    for (int k0 = 0; k0 < DFEAT; k0 += 4) {
        v2f aA, aH;
        aA.x = Sa[m * 132 + k0 + koff];  aA.y = Sa[m * 132 + k0 + koff + 1];
        aH.x = Sh[m * 132 + k0 + koff];  aH.y = Sh[m * 132 + k0 + koff + 1];
        v2f bl = Bl2[(size_t)((k0 >> 1) + kh) * 128 + n];   // one b64 load
        v2f br = Br2[(size_t)((k0 >> 1) + kh) * 128 + n];
        acc = __builtin_amdgcn_wmma_f32_16x16x4_f32(false, aA, false, bl, (short)0, acc, false, false);
        acc = __builtin_amdgcn_wmma_f32_16x16x4_f32(false, aH, false, br, (short)0, acc, false, false);
    }
    float bv = bias[n];
    int mbase = (lane < 16) ? 0 : 8;
    #pragma unroll
    for (int r = 0; r < 8; ++r)
        out[(row0 + mbase + r) * HDIM + n] = acc[r] + bv;
}

// logits[16-row strip][64] += tmp @ post slice   (4 waves x 16-col tiles)
// Bp points into the packed post weights at this slice's K-pair offset.
__global__ void __launch_bounds__(128)
gemm_post_kernel(const float* __restrict__ A, const float* __restrict__ Bp,
                 float* __restrict__ logits) {
    __shared__ float S[16 * 132];
    int tid = threadIdx.x;
    long row0 = (long)blockIdx.x * 16;
    #pragma unroll
    for (int j = 0; j < 16; ++j) {
        int i = tid + j * 128;
        int r = i >> 7, c = i & 127;
        S[r * 132 + c] = A[(row0 + r) * DFEAT + c];
    }
    __syncthreads();

    int lane = tid & 31, wave = tid >> 5;
    int m    = lane & 15;
    int koff = (lane < 16) ? 0 : 2;
    int kh   = (lane < 16) ? 0 : 1;
    int n    = wave * 16 + m;
    const v2f* B2 = (const v2f*)Bp;              // float2 index: p*64 + n

    v8f acc = {};
    #pragma unroll 4
    for (int k0 = 0; k0 < DFEAT; k0 += 4) {
        v2f a;
        a.x = S[m * 132 + k0 + koff];  a.y = S[m * 132 + k0 + koff + 1];
        v2f b = B2[(size_t)((k0 >> 1) + kh) * 64 + n];
        acc = __builtin_amdgcn_wmma_f32_16x16x4_f32(false, a, false, b, (short)0, acc, false, false);
    }
    int mbase = (lane < 16) ? 0 : 8;
    #pragma unroll
    for (int r = 0; r < 8; ++r) {
        long idx = (row0 + mbase + r) * NCLS + n;
        logits[idx] += acc[r];            // slices processed sequentially: race-free
    }
}

// ---------------- log_softmax over 64 classes, wave per row ----------------
__global__ void __launch_bounds__(256)
logsoftmax_kernel(float* __restrict__ out, int N) {
    int wave = threadIdx.x >> 5, lane = threadIdx.x & 31;
    int row = blockIdx.x * 8 + wave;
    if (row >= N) return;
    float v0 = out[(long)row * NCLS + lane];
    float v1 = out[(long)row * NCLS + lane + 32];
    float m = fmaxf(v0, v1);
    #pragma unroll
    for (int off = 16; off >= 1; off >>= 1) m = fmaxf(m, __shfl_xor(m, off, 32));
    float s = expf(v0 - m) + expf(v1 - m);
    #pragma unroll
    for (int off = 16; off >= 1; off >>= 1) s += __shfl_xor(s, off, 32);
    float ls = logf(s);
    out[(long)row * NCLS + lane]      = v0 - m - ls;
    out[(long)row * NCLS + lane + 32] = v1 - m - ls;
}

// ---------------------------------------------------------------------------

extern "C" void kernel_launch(void* const* d_in, const int* in_sizes, int n_in,
                              void* d_out, int out_size, void* d_ws, size_t ws_size,
                              hipStream_t stream) {
    const float* x       = (const float*)d_in[0];
    const int*   ei0     = (const int*)d_in[1];
    const int*   ei1     = (const int*)d_in[2];
    const float* Wl_mean = (const float*)d_in[3];
    const float* bl_mean = (const float*)d_in[4];
    const float* Wr_mean = (const float*)d_in[5];
    const float* Wl_max  = (const float*)d_in[6];
    const float* bl_max  = (const float*)d_in[7];
    const float* Wr_max  = (const float*)d_in[8];
    const float* post_W  = (const float*)d_in[9];
    const float* post_b  = (const float*)d_in[10];

    const int N = in_sizes[0] / DFEAT;   // 40000 (divisible by 16)
    const int E = in_sizes[1] / 2;       // 640000

    // workspace: h_relu | agg | tmp | cnt | 8x packed W(128x128) | packed post(512x64)
    float* ws     = (float*)d_ws;
    float* h_relu = ws;
    float* agg    = h_relu + (size_t)N * DFEAT;
    float* tmp    = agg    + (size_t)N * DFEAT;
    float* cnt    = tmp    + (size_t)N * DFEAT;
    float* WP     = cnt + N;
    float* postP  = WP + (size_t)8 * 128 * 128;
    float* logits = (float*)d_out;

    // pack weights: order per layer: [Wl_mean, Wr_mean, Wl_max, Wr_max]
    const float* wsrc[8] = { Wl_mean,          Wr_mean,          Wl_max,          Wr_max,
                             Wl_mean + 16384,  Wr_mean + 16384,  Wl_max + 16384,  Wr_max + 16384 };
    for (int i = 0; i < 8; ++i)
        pack_w128_kernel<<<64, 256, 0, stream>>>(wsrc[i], WP + (size_t)i * 16384);
    pack_post_kernel<<<128, 256, 0, stream>>>(post_W, postP);

    relu_kernel<<<(N * DFEAT + 255) / 256, 256, 0, stream>>>(x, h_relu, N * DFEAT);
    init_logits_kernel<<<(N * NCLS + 255) / 256, 256, 0, stream>>>(logits, post_b, N * NCLS);

    for (int layer = 0; layer < 2; ++layer) {
        const float* h_in = (layer == 0) ? x : h_relu;   // ref: h is relu(x) for layer 1
        const int* ei  = (layer == 0) ? ei0 : ei1;
        const int* src = ei;
        const int* dst = ei + E;
        const float* WlP   = WP + (size_t)(layer * 4 + 0) * 16384;
        const float* WrP   = WP + (size_t)(layer * 4 + 1) * 16384;
        const float* WlPmx = WP + (size_t)(layer * 4 + 2) * 16384;
        const float* WrPmx = WP + (size_t)(layer * 4 + 3) * 16384;

        // ---- mean-aggr conv -> post slice (2*layer) ----
        hipMemsetAsync(agg, 0, (size_t)N * DFEAT * sizeof(float), stream);
        hipMemsetAsync(cnt, 0, (size_t)N * sizeof(float), stream);
        scatter_mean_kernel<<<SCATTER_BLOCKS, 256, 0, stream>>>(src, dst, h_in, agg, cnt, E);
        mean_div_kernel<<<(N * DFEAT + 255) / 256, 256, 0, stream>>>(agg, cnt, N * DFEAT);
        gemm_conv_kernel<<<N / 16, 256, 0, stream>>>(agg, h_in, WlP, WrP,
                                                     bl_mean + layer * HDIM, tmp);
        gemm_post_kernel<<<N / 16, 128, 0, stream>>>(tmp,
                postP + (size_t)(2 * layer + 0) * 64 * 128, logits);

        // ---- max-aggr conv -> post slice (2*layer+1) ----
        fill_u32_kernel<<<(N * DFEAT + 255) / 256, 256, 0, stream>>>(
                (unsigned*)agg, ORD_NEG_INF, N * DFEAT);
        scatter_max_kernel<<<SCATTER_BLOCKS, 256, 0, stream>>>(src, dst, h_in, (unsigned*)agg, E);
        max_decode_kernel<<<(N * DFEAT + 255) / 256, 256, 0, stream>>>(
                (const unsigned*)agg, agg, N * DFEAT);
        gemm_conv_kernel<<<N / 16, 256, 0, stream>>>(agg, h_in, WlPmx, WrPmx,
                                                     bl_max + layer * HDIM, tmp);
        gemm_post_kernel<<<N / 16, 128, 0, stream>>>(tmp,
                postP + (size_t)(2 * layer + 1) * 64 * 128, logits);
    }

    logsoftmax_kernel<<<(N + 7) / 8, 256, 0, stream>>>(logits, N);
}